// capsule_fusion_20675972563756
// MI455X (gfx1250) — compile-verified
//
#include <hip/hip_runtime.h>
#include <cmath>

// MI455X / gfx1250, wave32. Matrix ops via V_WMMA_F32_16X16X4_F32 (full f32;
// problem is bandwidth-bound: ~3 GFLOP vs ~200 MB HBM traffic, V stays in L2).
typedef float v2f __attribute__((ext_vector_type(2)));
typedef float v8f __attribute__((ext_vector_type(8)));

#define EPSF 1e-8f
#define Ncap 64     // batch
#define Lcap 128    // input capsules
#define Dcap 768    // feature dim
#define IND  64
#define OUTD 64
#define NOUT 32

// ---------------------------------------------------------------------------
// K1: f_a[l,n] = sigmoid( x_t[l,n,:] . Wscore[l,:] + Bscore[l] )
// one wave per (l,n)
// ---------------------------------------------------------------------------
__global__ void __launch_bounds__(256) k_score(const float* __restrict__ x,
                                               const float* __restrict__ Wscore,
                                               const float* __restrict__ Bscore,
                                               float* __restrict__ f_a) {
  int wid  = (blockIdx.x * 256 + threadIdx.x) >> 5;   // 0..8191 == l*64+n
  int lane = threadIdx.x & 31;
  int l = wid >> 6, n = wid & 63;
  const float* xr = x + (size_t)n * (Lcap * Dcap) + (size_t)l * Dcap;
  const float* w  = Wscore + (size_t)l * Dcap;
  float s = 0.f;
  #pragma unroll 4
  for (int d = lane; d < Dcap; d += 32) s += xr[d] * w[d];
  #pragma unroll
  for (int off = 16; off; off >>= 1) s += __shfl_xor(s, off, 32);
  if (lane == 0) {
    float a = s + Bscore[l];
    f_a[wid] = 1.0f / (1.0f + expf(-a));
  }
}

// ---------------------------------------------------------------------------
// K2: mu_in[l,n,i] = gelu( x_t[l,n,:] @ Wcap[l,:,i] + Bcap[l,i] )
// one block per l; 64x64 C; K=768 chunked by 32; 8 waves x 2 WMMA tiles.
// Both A and B tiles stored row(M/N)-major, K-minor in LDS so each lane's
// (K,K+1) fragment pair is one aligned b64 load. Row stride 36 floats:
// 144 B rows (16B aligned for float4 staging), 36r mod 64 distinct banks.
// ---------------------------------------------------------------------------
#define STR2 36
__global__ void __launch_bounds__(256) k_mu(const float* __restrict__ x,
                                            const float* __restrict__ Wcap,
                                            const float* __restrict__ Bcap,
                                            float* __restrict__ mu) {
  __shared__ float Xs [64 * STR2];   // [n][kk]  kk<32
  __shared__ float WsT[64 * STR2];   // [i][kk]  kk<32  (transposed B tile)
  const int l    = blockIdx.x;
  const int tid  = threadIdx.x;
  const int wave = tid >> 5, lane = tid & 31;
  const int t0 = wave * 2, t1 = t0 + 1;
  const int mi0 = t0 >> 2, ni0 = t0 & 3;
  const int mi1 = t1 >> 2, ni1 = t1 & 3;
  const int row = lane & 15;
  const int kbo = (lane >> 4) << 1;      // 0 or 2 : K pair owned by this lane
  v8f c0 = {}; v8f c1 = {};

  for (int kc = 0; kc < Dcap; kc += 32) {
    // stage X chunk: 64 x 32, float4 global loads (b128), aligned LDS stores
    #pragma unroll
    for (int e = 0; e < 2; e++) {
      int idx = tid + e * 256;                 // 0..511 float4s
      int n = idx >> 3, u = idx & 7;
      *(float4*)&Xs[n * STR2 + u * 4] =
          *(const float4*)&x[(size_t)n * (Lcap * Dcap) + (size_t)l * Dcap + kc + u * 4];
    }
    // stage W chunk transposed: read [kk][i] rows as float4, scatter to [i][kk]
    #pragma unroll
    for (int e = 0; e < 2; e++) {
      int idx = tid + e * 256;                 // 0..511 float4s
      int kk = idx >> 4, q = idx & 15;
      float4 g = *(const float4*)&Wcap[(size_t)l * (Dcap * IND) + (size_t)(kc + kk) * IND + q * 4];
      WsT[(q * 4 + 0) * STR2 + kk] = g.x;
      WsT[(q * 4 + 1) * STR2 + kk] = g.y;
      WsT[(q * 4 + 2) * STR2 + kk] = g.z;
      WsT[(q * 4 + 3) * STR2 + kk] = g.w;
    }
    if (kc + 32 < Dcap)   // prefetch next W chunk (global_prefetch_b8)
      __builtin_prefetch(&Wcap[(size_t)l * (Dcap * IND) + (size_t)(kc + 32) * IND + tid * 8], 0, 1);
    __syncthreads();

    #pragma unroll
    for (int k = 0; k < 32; k += 4) {
      int kk = k + kbo;
      v2f a0 = *(const v2f*)&Xs [(mi0 * 16 + row) * STR2 + kk];
      v2f b0 = *(const v2f*)&WsT[(ni0 * 16 + row) * STR2 + kk];
      c0 = __builtin_amdgcn_wmma_f32_16x16x4_f32(false, a0, false, b0,
                                                 (short)0, c0, false, false);
      v2f a1 = *(const v2f*)&Xs [(mi1 * 16 + row) * STR2 + kk];
      v2f b1 = *(const v2f*)&WsT[(ni1 * 16 + row) * STR2 + kk];
      c1 = __builtin_amdgcn_wmma_f32_16x16x4_f32(false, a1, false, b1,
                                                 (short)0, c1, false, false);
    }
    __syncthreads();
  }

  const int mofs = (lane >> 4) * 8;   // C/D: M = r + 8*(lane>=16), N = lane&15
  #pragma unroll
  for (int r = 0; r < 8; r++) {
    {
      int n = mi0 * 16 + r + mofs, i = ni0 * 16 + row;
      float v = c0[r] + Bcap[l * IND + i];
      mu[((size_t)l * Ncap + n) * IND + i] = 0.5f * v * (1.0f + erff(v * 0.70710678118654752f));
    }
    {
      int n = mi1 * 16 + r + mofs, i = ni1 * 16 + row;
      float v = c1[r] + Bcap[l * IND + i];
      mu[((size_t)l * Ncap + n) * IND + i] = 0.5f * v * (1.0f + erff(v * 0.70710678118654752f));
    }
  }
}

// ---------------------------------------------------------------------------
// K3: V[l,o,n,j] = mu[l,n,:] @ Wvote[l,o,:,j] + Bvote[l,o,j]
// one block per (l,o); 64x64x64 GEMM; same b64-fragment LDS layout.
// Row stride 68 floats: 272 B rows (16B aligned), 68r mod 64 = 4r distinct.
// ---------------------------------------------------------------------------
#define STR3 68
__global__ void __launch_bounds__(256) k_votes(const float* __restrict__ mu,
                                               const float* __restrict__ Wvote,
                                               const float* __restrict__ Bvote,
                                               float* __restrict__ V) {
  __shared__ float As [64 * STR3];   // [n][i]
  __shared__ float BsT[64 * STR3];   // [j][i]  (transposed B tile)
  const int lo   = blockIdx.x;       // l*32 + o
  const int tid  = threadIdx.x;
  const int wave = tid >> 5, lane = tid & 31;
  const int t0 = wave * 2, t1 = t0 + 1;
  const int mi0 = t0 >> 2, ni0 = t0 & 3;
  const int mi1 = t1 >> 2, ni1 = t1 & 3;
  const int row = lane & 15;
  const int kbo = (lane >> 4) << 1;
  const int l = lo >> 5;
  v8f c0 = {}; v8f c1 = {};

  // stage A (mu tile): 4096 floats = 1024 float4s
  #pragma unroll
  for (int e = 0; e < 4; e++) {
    int idx = tid + e * 256;
    int n = idx >> 4, q = idx & 15;
    *(float4*)&As[n * STR3 + q * 4] =
        *(const float4*)&mu[(size_t)l * (Ncap * IND) + n * IND + q * 4];
  }
  // stage B transposed: read Wvote[lo] rows [i][j] as float4, scatter to [j][i]
  #pragma unroll
  for (int e = 0; e < 4; e++) {
    int idx = tid + e * 256;
    int i = idx >> 4, q = idx & 15;
    float4 g = *(const float4*)&Wvote[(size_t)lo * (IND * OUTD) + i * OUTD + q * 4];
    BsT[(q * 4 + 0) * STR3 + i] = g.x;
    BsT[(q * 4 + 1) * STR3 + i] = g.y;
    BsT[(q * 4 + 2) * STR3 + i] = g.z;
    BsT[(q * 4 + 3) * STR3 + i] = g.w;
  }
  __syncthreads();

  #pragma unroll
  for (int k = 0; k < 64; k += 4) {
    int kk = k + kbo;
    v2f a0 = *(const v2f*)&As [(mi0 * 16 + row) * STR3 + kk];
    v2f b0 = *(const v2f*)&BsT[(ni0 * 16 + row) * STR3 + kk];
    c0 = __builtin_amdgcn_wmma_f32_16x16x4_f32(false, a0, false, b0,
                                               (short)0, c0, false, false);
    v2f a1 = *(const v2f*)&As [(mi1 * 16 + row) * STR3 + kk];
    v2f b1 = *(const v2f*)&BsT[(ni1 * 16 + row) * STR3 + kk];
    c1 = __builtin_amdgcn_wmma_f32_16x16x4_f32(false, a1, false, b1,
                                               (short)0, c1, false, false);
  }

  const int mofs = (lane >> 4) * 8;
  #pragma unroll
  for (int r = 0; r < 8; r++) {
    {
      int n = mi0 * 16 + r + mofs, j = ni0 * 16 + row;
      V[((size_t)lo * Ncap + n) * OUTD + j] = c0[r] + Bvote[lo * OUTD + j];
    }
    {
      int n = mi1 * 16 + r + mofs, j = ni1 * 16 + row;
      V[((size_t)lo * Ncap + n) * OUTD + j] = c1[r] + Bvote[lo * OUTD + j];
    }
  }
}

// ---------------------------------------------------------------------------
// K4: EM M-step. One block per (o,n); threads = j (64). Serial over l.
// Single pass: var = sum(D*V^2)/denom - mu^2  (exact identity, clamped >= 0)
// R == nullptr -> uniform R = 1/32 (first iteration)
// ---------------------------------------------------------------------------
__global__ void __launch_bounds__(64) k_em(const float* __restrict__ V,
                                           const float* __restrict__ f_a,
                                           const float* __restrict__ R,
                                           const float* __restrict__ beta_use,
                                           const float* __restrict__ beta_ign,
                                           float* __restrict__ a_out,
                                           float* __restrict__ mu_out,
                                           float* __restrict__ var_out) {
  const int on = blockIdx.x;          // o*64 + n
  const int o = on >> 6, n = on & 63;
  const int j = threadIdx.x;
  const float bi = beta_ign[o];
  float sD = 0.f, sDV = 0.f, sDV2 = 0.f, aAcc = 0.f;
  for (int l = 0; l < Lcap; l++) {
    float fa = f_a[l * Ncap + n];
    float r  = R ? R[(l * NOUT + o) * Ncap + n] : (1.0f / 32.0f);
    float Du = fa * r;
    float v  = V[((size_t)l * (NOUT * Ncap) + on) * OUTD + j];
    sD   += Du;
    sDV  += Du * v;
    sDV2 += Du * v * v;
    aAcc += beta_use[l] * Du - bi * (fa - Du);
  }
  float denom = sD + EPSF;
  float m   = sDV / denom;
  float var = sDV2 / denom - m * m;
  var = fmaxf(var, 0.0f);
  mu_out[on * OUTD + j]  = m;
  var_out[on * OUTD + j] = var;
  if (j == 0) a_out[on] = aAcc;
}

// ---------------------------------------------------------------------------
// K5: E-step routing. One wave per (l,n). Lanes cover j = {lane, lane+32};
// per-o log-likelihood reduced across the wave; softmax over o via shuffles.
// ---------------------------------------------------------------------------
__global__ void __launch_bounds__(256) k_R(const float* __restrict__ V,
                                           const float* __restrict__ a_out,
                                           const float* __restrict__ mu_out,
                                           const float* __restrict__ var_out,
                                           float* __restrict__ R) {
  const int wid  = (blockIdx.x * 256 + threadIdx.x) >> 5;   // l*64 + n
  const int lane = threadIdx.x & 31;
  const int l = wid >> 6, n = wid & 63;
  float myT = 0.f;
  for (int o = 0; o < NOUT; o++) {
    int on = o * Ncap + n;
    float s1 = 0.f, s2 = 0.f;
    #pragma unroll
    for (int jj = 0; jj < 2; jj++) {
      int j = lane + jj * 32;
      float v  = V[((size_t)l * (NOUT * Ncap) + on) * OUTD + j];
      float m  = mu_out[on * OUTD + j];
      float vr = var_out[on * OUTD + j];
      float d  = v - m;
      s1 += d * d / (2.0f * vr + EPSF);
      s2 += logf(vr + EPSF);
    }
    #pragma unroll
    for (int off = 16; off; off >>= 1) {
      s1 += __shfl_xor(s1, off, 32);
      s2 += __shfl_xor(s2, off, 32);
    }
    float logp = -s1 - 1.0f - 0.5f * 3.14159265358979323846f - 0.5f * s2;
    float a = a_out[on];
    float logsig = fminf(a, 0.0f) - log1pf(expf(-fabsf(a)));
    float t = logsig + logp;
    if (lane == o) myT = t;
  }
  // softmax over the 32 output capsules (one per lane)
  float mx = myT;
  #pragma unroll
  for (int off = 16; off; off >>= 1) mx = fmaxf(mx, __shfl_xor(mx, off, 32));
  float e = expf(myT - mx);
  float s = e;
  #pragma unroll
  for (int off = 16; off; off >>= 1) s += __shfl_xor(s, off, 32);
  R[(l * NOUT + lane) * Ncap + n] = e / s;
}

// ---------------------------------------------------------------------------
// K6: transpose to output layout: out = [ a_out^T (N,32) | mu_out^T (N,32,64) ]
// ---------------------------------------------------------------------------
__global__ void __launch_bounds__(256) k_final(const float* __restrict__ a_out,
                                               const float* __restrict__ mu_out,
                                               float* __restrict__ out) {
  int i = blockIdx.x * 256 + threadIdx.x;
  if (i < Ncap * NOUT) {               // out[n*32+o] = a_out[o*64+n]
    int n = i >> 5, o = i & 31;
    out[i] = a_out[o * Ncap + n];
  }
  if (i < Ncap * NOUT * OUTD) {        // out[2048 + (n*32+o)*64+j] = mu_out[(o*64+n)*64+j]
    int j = i & 63, no = i >> 6;
    int o = no & 31, n = no >> 5;
    out[Ncap * NOUT + i] = mu_out[(o * Ncap + n) * OUTD + j];
  }
}

// ---------------------------------------------------------------------------
extern "C" void kernel_launch(void* const* d_in, const int* in_sizes, int n_in,
                              void* d_out, int out_size, void* d_ws, size_t ws_size,
                              hipStream_t stream) {
  const float* x        = (const float*)d_in[0];
  const float* Wscore   = (const float*)d_in[1];
  const float* Bscore   = (const float*)d_in[2];
  const float* Wcap     = (const float*)d_in[3];
  const float* Bcap     = (const float*)d_in[4];
  const float* Wvote    = (const float*)d_in[5];
  const float* Bvote    = (const float*)d_in[6];
  const float* beta_use = (const float*)d_in[7];
  const float* beta_ign = (const float*)d_in[8];
  // d_in[9] = iters (== 3 per reference setup; launch sequence hardcoded)
  float* out = (float*)d_out;

  // workspace partition (floats)
  float* ws = (float*)d_ws;
  float* f_a     = ws;                       // 8192
  float* mu      = f_a + 8192;               // 524288
  float* V       = mu + 524288;              // 16777216
  float* a_out   = V + 16777216;             // 2048
  float* mu_out  = a_out + 2048;             // 131072
  float* var_out = mu_out + 131072;          // 131072
  float* R       = var_out + 131072;         // 262144

  k_score <<<1024, 256, 0, stream>>>(x, Wscore, Bscore, f_a);
  k_mu    <<<Lcap, 256, 0, stream>>>(x, Wcap, Bcap, mu);
  k_votes <<<Lcap * NOUT, 256, 0, stream>>>(mu, Wvote, Bvote, V);

  // EM: iters = 3  ->  em_step, then 2x (E-step routing + em_step)
  k_em <<<NOUT * Ncap, 64, 0, stream>>>(V, f_a, nullptr, beta_use, beta_ign,
                                        a_out, mu_out, var_out);
  for (int it = 0; it < 2; it++) {
    k_R  <<<1024, 256, 0, stream>>>(V, a_out, mu_out, var_out, R);
    k_em <<<NOUT * Ncap, 64, 0, stream>>>(V, f_a, R, beta_use, beta_ign,
                                          a_out, mu_out, var_out);
  }

  k_final <<<(Ncap * NOUT * OUTD + 255) / 256, 256, 0, stream>>>(a_out, mu_out, out);
}